// MaskedFastDiffusionModel_32530082300175
// MI455X (gfx1250) — compile-verified
//
#include <hip/hip_runtime.h>
#include <hip/hip_bf16.h>
#include <math.h>

// Problem constants (from reference setup_inputs)
#define BB 4
#define SS 1024
#define VV 50257
#define DD 768
#define HH 384
#define ROWS (BB * SS)          // 4096
#define THRESH 0.7f

typedef float v2f __attribute__((ext_vector_type(2)));
typedef float v8f __attribute__((ext_vector_type(8)));

// ---------------------------------------------------------------------------
// Kernel 1: per-row online softmax stats (1/sum(exp(x-max)) == exp(max-lse))
//           fused with token argmax (optionally Gumbel-perturbed).
// One 256-thread workgroup per row; 128-bit streaming loads of logits.
// ---------------------------------------------------------------------------
__device__ __forceinline__ float gumbel_hash(unsigned int a) {
    a ^= a >> 16; a *= 0x7feb352du; a ^= a >> 15; a *= 0x846ca68bu; a ^= a >> 16;
    float u = (float)(a >> 8) * (1.0f / 16777216.0f) + 1e-20f;
    return -__logf(-__logf(u));
}

__global__ __launch_bounds__(256)
void softmax_argmax_kernel(const float* __restrict__ logits,
                           const int* __restrict__ step_p,
                           const int* __restrict__ total_p,
                           float* __restrict__ maxprob,   // [ROWS]
                           int* __restrict__ token)       // [ROWS]
{
    const int row = blockIdx.x;
    const int t   = threadIdx.x;
    const bool useG = (step_p[0] < (total_p[0] >> 1));

    const size_t rbase = (size_t)row * VV;
    const float* x = logits + rbase;

    float m = -INFINITY, s = 0.0f;       // online softmax state
    float gm = -INFINITY; int gidx = 0;  // argmax state

    // alignment peel: make float4 region 16B aligned
    const int mis  = (int)(rbase & 3u);
    const int head = (4 - mis) & 3;

    if (t < head) {
        float val = x[t];
        float av = val + (useG ? gumbel_hash((unsigned)(row * 50423 + t)) : 0.0f);
        gm = av; gidx = t;
        m = val; s = 1.0f;
    }

    const int count4 = (VV - head) >> 2;
    const float4* x4 = (const float4*)(x + head);
    for (int i = t; i < count4; i += 256) {
        float4 v = x4[i];
        int e0 = head + 4 * i;
        float vv[4] = {v.x, v.y, v.z, v.w};
#pragma unroll
        for (int j = 0; j < 4; ++j) {
            float val = vv[j];
            float nm = fmaxf(m, val);
            s = s * __expf(m - nm) + __expf(val - nm);
            m = nm;
            float av = val + (useG ? gumbel_hash((unsigned)(row * 50423 + e0 + j)) : 0.0f);
            if (av > gm) { gm = av; gidx = e0 + j; }
        }
    }

    // tail
    int tidx = head + 4 * count4 + t;
    if (tidx < VV) {
        float val = x[tidx];
        float nm = fmaxf(m, val);
        s = s * __expf(m - nm) + __expf(val - nm);
        m = nm;
        float av = val + (useG ? gumbel_hash((unsigned)(row * 50423 + tidx)) : 0.0f);
        if (av > gm) { gm = av; gidx = tidx; }
    }

    // workgroup reduction
    __shared__ float sm[256], ssum[256], sgm[256];
    __shared__ int   sgi[256];
    sm[t] = m; ssum[t] = s; sgm[t] = gm; sgi[t] = gidx;
    __syncthreads();
    for (int o = 128; o > 0; o >>= 1) {
        if (t < o) {
            float m2 = sm[t + o], s2 = ssum[t + o];
            float nm = fmaxf(sm[t], m2);
            ssum[t] = ssum[t] * __expf(sm[t] - nm) + s2 * __expf(m2 - nm);
            sm[t] = nm;
            // argmax: prefer lower index on ties (strict >)
            if (sgm[t + o] > sgm[t] ||
                (sgm[t + o] == sgm[t] && sgi[t + o] < sgi[t])) {
                sgm[t] = sgm[t + o]; sgi[t] = sgi[t + o];
            }
        }
        __syncthreads();
    }
    if (t == 0) {
        maxprob[row] = 1.0f / ssum[0];   // exp(max - logsumexp)
        token[row]   = sgi[0];
    }
}

// ---------------------------------------------------------------------------
// Kernel 2: fused MLP head via V_WMMA_F32_16X16X4_F32.
//   pre = H[16x768] @ W1[768x384] + b1 ; g = gelu(pre)
//   learned = sigmoid(g @ w2 + b2) ; conf = (0.8*maxprob + 0.2*learned)*mask
// One 256-thread (8-wave) workgroup per 16-row M block. A staged in LDS.
// Each wave owns 3 contiguous 16x16 N-tiles (8 waves * 48 = 384 = full N).
// ---------------------------------------------------------------------------
__global__ __launch_bounds__(256)
void mlp_conf_kernel(const float* __restrict__ Hs,     // [ROWS, DD]
                     const float* __restrict__ W1,     // [DD, HH]
                     const float* __restrict__ b1,     // [HH]
                     const float* __restrict__ w2,     // [HH]
                     const float* __restrict__ b2,     // [1]
                     const int*   __restrict__ mask,   // [ROWS]
                     const float* __restrict__ maxprob,// [ROWS]
                     float* __restrict__ conf_out)     // [ROWS] (d_out slice 0)
{
    const int m0   = blockIdx.x * 16;       // row block
    const int t    = threadIdx.x;
    const int wave = t >> 5;                // 0..7
    const int lane = t & 31;
    const int lhlf = lane >> 4;             // 0 or 1 (half-wave)
    const int l15  = lane & 15;

    __shared__ float As[16 * 64];           // A chunk: 16 rows x 64 k
    __shared__ float Gl[16 * HH];           // gelu activations, 16 x 384
    __shared__ float Rp[256];               // row-dot partials

    v8f acc[3] = {v8f{}, v8f{}, v8f{}};

    for (int kc = 0; kc < DD; kc += 64) {
        // stage 16x64 A chunk (each thread one float4)
        {
            int r = t >> 4, c4 = (t & 15) << 2;
            const float4 v = *(const float4*)(Hs + (size_t)(m0 + r) * DD + kc + c4);
            *(float4*)(&As[r * 64 + c4]) = v;
        }
        __syncthreads();

        for (int k = 0; k < 64; k += 4) {
            // A fragment 16x4: lanes 0-15 -> K=k,k+1 ; lanes 16-31 -> K=k+2,k+3
            const int ka = k + (lhlf << 1);
            v2f a;
            a.x = As[l15 * 64 + ka];
            a.y = As[l15 * 64 + ka + 1];
#pragma unroll
            for (int j = 0; j < 3; ++j) {
                const int n0 = wave * 48 + j * 16;
                const int kb = kc + k + (lhlf << 1);     // global K for B frag
                v2f b;
                b.x = W1[(size_t)kb * HH + n0 + l15];
                b.y = W1[(size_t)(kb + 1) * HH + n0 + l15];
                acc[j] = __builtin_amdgcn_wmma_f32_16x16x4_f32(
                    /*neg_a=*/false, a, /*neg_b=*/false, b,
                    /*c_mod=*/(short)0, acc[j],
                    /*reuse_a=*/false, /*reuse_b=*/false);
            }
        }
        __syncthreads();
    }

    // bias + exact GELU -> LDS (C/D layout: vgpr r holds M=r (lanes 0-15) or
    // M=r+8 (lanes 16-31); N = n0 + lane&15)
#pragma unroll
    for (int j = 0; j < 3; ++j) {
        const int n = wave * 48 + j * 16 + l15;
        const float bias = b1[n];
#pragma unroll
        for (int r = 0; r < 8; ++r) {
            const int mrow = r + (lhlf << 3);
            float v = acc[j][r] + bias;
            float g = 0.5f * v * (1.0f + erff(v * 0.70710678118654752f));
            Gl[mrow * HH + n] = g;
        }
    }
    __syncthreads();

    // learned head: 16 threads per row, each sums 24 strided columns
    {
        const int row = t >> 4, sub = t & 15;
        float p = 0.0f;
        for (int n = sub; n < HH; n += 16) p += Gl[row * HH + n] * w2[n];
        Rp[t] = p;
    }
    __syncthreads();
    if (t < 16) {
        float sum = 0.0f;
#pragma unroll
        for (int i = 0; i < 16; ++i) sum += Rp[t * 16 + i];
        sum += b2[0];
        const float learned = 1.0f / (1.0f + __expf(-sum));
        const int g = m0 + t;
        float c = 0.8f * maxprob[g] + 0.2f * learned;
        c *= (mask[g] != 0) ? 1.0f : 0.0f;
        conf_out[g] = c;
    }
}

// ---------------------------------------------------------------------------
// Kernel 3: per-batch decision. One 1024-thread block per batch row.
// ---------------------------------------------------------------------------
__global__ __launch_bounds__(1024)
void decide_kernel(const float* __restrict__ conf,     // [ROWS]
                   const int*   __restrict__ mask,     // [ROWS]
                   const int*   __restrict__ token,    // [ROWS]
                   float* __restrict__ out_newmask,    // [ROWS]
                   int*   __restrict__ out_tokens)     // [ROWS]
{
    const int b = blockIdx.x;
    const int s = threadIdx.x;
    const int idx = b * SS + s;

    const float c  = conf[idx];
    const bool  mk = (mask[idx] != 0);
    const bool  above = mk && (c > THRESH);
    const float mc = mk ? c : -INFINITY;

    __shared__ float vals[SS];
    __shared__ int   idxs[SS];
    __shared__ int   anyA, anyM;
    if (s == 0) { anyA = 0; anyM = 0; }
    vals[s] = mc; idxs[s] = s;
    __syncthreads();
    if (above) atomicOr(&anyA, 1);
    if (mk)    atomicOr(&anyM, 1);
    for (int o = SS / 2; o > 0; o >>= 1) {
        if (s < o) {
            if (vals[s + o] > vals[s]) {           // strict >: first index wins ties
                vals[s] = vals[s + o]; idxs[s] = idxs[s + o];
            }
        }
        __syncthreads();
    }
    const int  best = idxs[0];
    const bool anyAbove = (anyA != 0);
    const bool hasM = (anyM != 0);

    const bool fallback = (s == best) && hasM;
    const bool unmask = (anyAbove ? above : fallback) && hasM;
    const bool nmask = mk && !unmask;

    out_newmask[idx] = nmask ? 1.0f : 0.0f;
    out_tokens[idx]  = unmask ? token[idx] : 0;
}

// ---------------------------------------------------------------------------
extern "C" void kernel_launch(void* const* d_in, const int* in_sizes, int n_in,
                              void* d_out, int out_size, void* d_ws, size_t ws_size,
                              hipStream_t stream) {
    const float* logits  = (const float*)d_in[0];
    const float* hidden  = (const float*)d_in[1];
    const int*   mask    = (const int*)d_in[2];
    const float* w1      = (const float*)d_in[3];
    const float* b1      = (const float*)d_in[4];
    const float* w2      = (const float*)d_in[5];
    const float* b2      = (const float*)d_in[6];
    const int*   step    = (const int*)d_in[7];
    const int*   totalst = (const int*)d_in[8];

    // workspace: maxprob f32[ROWS] | tokens i32[ROWS]
    float* ws_maxprob = (float*)d_ws;
    int*   ws_token   = (int*)((char*)d_ws + ROWS * sizeof(float));

    // output: conf f32[ROWS] | new_mask f32[ROWS] | tokens i32[ROWS]
    float* out_conf    = (float*)d_out;
    float* out_newmask = out_conf + ROWS;
    int*   out_tokens  = (int*)((char*)d_out + 2 * ROWS * sizeof(float));

    softmax_argmax_kernel<<<ROWS, 256, 0, stream>>>(
        logits, step, totalst, ws_maxprob, ws_token);

    mlp_conf_kernel<<<ROWS / 16, 256, 0, stream>>>(
        hidden, w1, b1, w2, b2, mask, ws_maxprob, out_conf);

    decide_kernel<<<BB, SS, 0, stream>>>(
        out_conf, mask, ws_token, out_newmask, out_tokens);
}